// CentroidGATConv_83330955477531
// MI455X (gfx1250) — compile-verified
//
#include <hip/hip_runtime.h>
#include <hip/hip_bf16.h>
#include <math.h>

// ---------------------------------------------------------------------------
// GAT forward for MI455X (gfx1250, wave32).
//   K1: ft = feat @ W          -- f32 WMMA 16x16x4, 32 rows x 256 cols / block
//                                 (2 row-tiles x 4 col-tiles per wave: B reuse
//                                 without register spills), NT feat loads
//   K2: el/er = <ft, attn_l/r> -- one wave per (node, head), float2 + shfl
//   K3: CSR starts[] from sorted dst via binary search
//   K4: segment softmax + weighted aggregate; ft gather hits 192MB L2,
//       float2 lanes, NT stores for write-once output
// ---------------------------------------------------------------------------

typedef float v2f __attribute__((ext_vector_type(2)));
typedef float v8f __attribute__((ext_vector_type(8)));

#define DIN   256   // D_in == H*D_out == 256 for this problem
#define HD    256
#define HEADS 4
#define DOUT  64
#define NEG_SLOPE 0.2f

// ---- Kernel 1: ft = feat @ W via V_WMMA_F32_16X16X4_F32 -------------------
// Block: 128 threads (4 waves). Wave w covers cols [64w, 64w+64) for rows
// [m0, m0+32): 2 row-tiles x 4 col-tiles of 16x16 per wave (8 accumulators =
// 64 VGPRs: fits without spilling, amortizes each B fragment over 2 tiles).
// A frag (16x4 f32): lanes 0-15 row M=lane K=k0+0..1; lanes 16-31 row M=lane-16
// K=k0+2..3.  B frag (4x16): lanes 0-15 col N=lane K=k0+0..1; lanes 16-31
// col N=lane-16 K=k0+2..3.  C/D: VGPR r -> (M=r | r+8, N=lane%16).
__global__ void gat_gemm_wmma(const float* __restrict__ feat,
                              const float* __restrict__ W,
                              float* __restrict__ ft, int N) {
    const int m0   = blockIdx.x * 32;
    const int wave = threadIdx.x >> 5;        // 0..3 -> 64-col slab (== head)
    const int lane = threadIdx.x & 31;
    const int lrow = lane & 15;
    const int hi   = lane >> 4;               // 0/1: K sub-pair selector
    const int n0   = wave * 64;

    // per-row-tile A pointers, rows clamped for the ragged last block
    const float* ap[2];
    #pragma unroll
    for (int t = 0; t < 2; ++t) {
        int r = m0 + 16 * t + lrow;
        if (r >= N) r = N - 1;
        ap[t] = feat + (size_t)r * DIN + hi * 2;
    }
    const float* bptr = W + (size_t)(hi * 2) * HD + n0 + lrow;

    v8f c[2][4] = {};                          // [row-tile][col-tile]

    #pragma unroll 2
    for (int k0 = 0; k0 < DIN; k0 += 4) {
        // speculative prefetch of the next block's rows (same K range)
        __builtin_prefetch(ap[0] + k0 + 32 * DIN, 0, 0);

        // B fragments: loaded once, reused across both row tiles (hot in L2)
        const float* bp = bptr + (size_t)k0 * HD;
        v2f b[4];
        #pragma unroll
        for (int u = 0; u < 4; ++u) {
            b[u].x = bp[16 * u];
            b[u].y = bp[HD + 16 * u];
        }

        #pragma unroll
        for (int t = 0; t < 2; ++t) {
            // feat is streamed exactly once -> non-temporal (keep W/ft in L2)
            v2f a = __builtin_nontemporal_load((const v2f*)(ap[t] + k0));
            #pragma unroll
            for (int u = 0; u < 4; ++u)
                c[t][u] = __builtin_amdgcn_wmma_f32_16x16x4_f32(
                    false, a, false, b[u], (short)0, c[t][u], false, false);
        }
    }

    // store D: row = m0 + 16*t + 8*hi + r, col = n0 + 16*u + lrow
    #pragma unroll
    for (int t = 0; t < 2; ++t) {
        float* orow = ft + (size_t)(m0 + 16 * t + 8 * hi) * HD + n0 + lrow;
        #pragma unroll
        for (int r = 0; r < 8; ++r) {
            int row = m0 + 16 * t + 8 * hi + r;
            if (row < N) {
                #pragma unroll
                for (int u = 0; u < 4; ++u)
                    orow[(size_t)r * HD + 16 * u] = c[t][u][r];
            }
        }
    }
}

// ---- Kernel 2: el[n,h] = <ft[n,h,:], attn_l[h,:]>, er likewise ------------
// One wave per (node, head): 32 lanes x float2, shuffle reduction.
__global__ void gat_el_er(const float* __restrict__ ft,
                          const float* __restrict__ attn_l,
                          const float* __restrict__ attn_r,
                          float* __restrict__ el, float* __restrict__ er,
                          int N) {
    const int gw   = (blockIdx.x * blockDim.x + threadIdx.x) >> 5;
    const int lane = threadIdx.x & 31;
    const int n = gw >> 2;
    const int h = gw & 3;
    if (n >= N) return;

    const v2f* f  = (const v2f*)(ft + (size_t)n * HD + h * DOUT);
    const v2f* al = (const v2f*)(attn_l + h * DOUT);
    const v2f* ar = (const v2f*)(attn_r + h * DOUT);

    v2f fv = f[lane], alv = al[lane], arv = ar[lane];
    float sl = fv.x * alv.x + fv.y * alv.y;
    float sr = fv.x * arv.x + fv.y * arv.y;
    #pragma unroll
    for (int off = 16; off > 0; off >>= 1) {
        sl += __shfl_xor(sl, off, 32);
        sr += __shfl_xor(sr, off, 32);
    }
    if (lane == 0) {
        el[n * HEADS + h] = sl;
        er[n * HEADS + h] = sr;
    }
}

// ---- Kernel 3: CSR row pointers from sorted dst ---------------------------
__global__ void gat_starts(const int* __restrict__ dst, int E,
                           int* __restrict__ starts, int N) {
    const int n = blockIdx.x * blockDim.x + threadIdx.x;
    if (n > N) return;
    int lo = 0, hi = E;                  // lower_bound: first i with dst[i] >= n
    while (lo < hi) {
        int mid = (lo + hi) >> 1;
        if (dst[mid] < n) lo = mid + 1; else hi = mid;
    }
    starts[n] = lo;
}

// ---- Kernel 4: softmax over incoming edges + weighted aggregate -----------
// One block per destination node; wave h handles head h, lane owns d=2L..2L+1.
__global__ void gat_aggr(const float* __restrict__ ft,
                         const float* __restrict__ el,
                         const float* __restrict__ er,
                         const int* __restrict__ src,
                         const int* __restrict__ starts,
                         float* __restrict__ out) {
    const int n    = blockIdx.x;
    const int h    = threadIdx.x >> 5;
    const int lane = threadIdx.x & 31;

    const int s = starts[n];
    const int e = starts[n + 1];
    v2f* o2 = (v2f*)(out + (size_t)n * HD + h * DOUT);
    if (s == e) {                         // no incoming edges -> zero row
        v2f z = {0.f, 0.f};
        __builtin_nontemporal_store(z, &o2[lane]);
        return;
    }

    const float ern = er[n * HEADS + h];

    // pass 1: lane-strided max of leaky_relu(el[src]+er[n]) over edges
    float m = -INFINITY;
    for (int j = s + lane; j < e; j += 32) {
        float x = el[src[j] * HEADS + h] + ern;
        x = x > 0.f ? x : NEG_SLOPE * x;
        m = fmaxf(m, x);
    }
    #pragma unroll
    for (int off = 16; off > 0; off >>= 1)
        m = fmaxf(m, __shfl_xor(m, off, 32));

    // pass 2: sequential edges; el load is wave-uniform (broadcast),
    // ft[src] row is one b64/lane (256B coalesced, hot in 192MB L2).
    float denom = 0.f;
    v2f acc = {0.f, 0.f};
    for (int j = s; j < e; ++j) {
        const int sj = src[j];
        float x = el[sj * HEADS + h] + ern;
        x = x > 0.f ? x : NEG_SLOPE * x;
        float w = __expf(x - m);
        denom += w;
        const v2f* f2 = (const v2f*)(ft + (size_t)sj * HD + h * DOUT);
        v2f fv = f2[lane];
        acc.x = fmaf(w, fv.x, acc.x);
        acc.y = fmaf(w, fv.y, acc.y);
    }
    const float inv = 1.0f / denom;
    v2f res = {acc.x * inv, acc.y * inv};
    __builtin_nontemporal_store(res, &o2[lane]);   // write-once output
}

// ---------------------------------------------------------------------------
extern "C" void kernel_launch(void* const* d_in, const int* in_sizes, int n_in,
                              void* d_out, int out_size, void* d_ws, size_t ws_size,
                              hipStream_t stream) {
    const float* feat   = (const float*)d_in[0];
    const int*   src    = (const int*)  d_in[1];
    const int*   dst    = (const int*)  d_in[2];
    const float* W      = (const float*)d_in[3];
    const float* attn_l = (const float*)d_in[4];
    const float* attn_r = (const float*)d_in[5];
    float*       out    = (float*)d_out;

    const int N = in_sizes[0] / DIN;   // 50000
    const int E = in_sizes[1];         // 800000

    // workspace carve: ft [N*256] f32, el [N*4], er [N*4], starts [N+1] i32
    float* ft     = (float*)d_ws;
    float* el     = ft + (size_t)N * HD;
    float* er     = el + (size_t)N * HEADS;
    int*   starts = (int*)(er + (size_t)N * HEADS);

    // K1: projection GEMM (f32 WMMA), 32 rows x 256 cols per 128-thread block
    gat_gemm_wmma<<<(N + 31) / 32, 128, 0, stream>>>(feat, W, ft, N);

    // K2: attention logits, one wave per (node, head)
    {
        int waves  = N * HEADS;
        int blocks = (waves * 32 + 255) / 256;
        gat_el_er<<<blocks, 256, 0, stream>>>(ft, attn_l, attn_r, el, er, N);
    }

    // K3: CSR starts
    gat_starts<<<(N + 1 + 255) / 256, 256, 0, stream>>>(dst, E, starts, N);

    // K4: softmax + aggregate, one block (4 waves = 4 heads) per node
    gat_aggr<<<N, 128, 0, stream>>>(ft, el, er, src, starts, out);
}